// DifferentiableStack_24902220382525
// MI455X (gfx1250) — compile-verified
//
#include <hip/hip_runtime.h>
#include <hip/hip_bf16.h>

// DifferentiableStack on MI455X (gfx1250).
//
// Key identity: the reference's output depends only on stack slot 63, which
// obeys h_t = a_t*h_{t-1} + c_t*x_t with a=(1-p)(1-o), c=p(1-o) (per b,t).
// We evaluate it as a chunked scan: per 16-step chunk,
//   out[k,d] = sum_j W[k][j]*x[t0+j,d] + P[k]*h_in[d]
// where W (16x16 lower triangular) and P (cumprod of a) depend only on
// (b, chunk). The 16x16 @ 16xD part runs on the matrix cores via
// V_WMMA_F32_16X16X4_F32 (K=16 in 4 chained WMMAs); the serial dependency
// shrinks from L=2048 FMAs to L/16=128 rank-1 carry fixups per wave.

typedef __attribute__((ext_vector_type(2))) float v2f;
typedef __attribute__((ext_vector_type(8))) float v8f;

static constexpr int BB = 16;     // batch
static constexpr int LL = 2048;   // sequence length
static constexpr int DD = 512;    // feature dim
static constexpr int CK = 16;     // chunk (time) size == WMMA M/N
static constexpr int NC = LL / CK;// 128 chunks

// ---------------------------------------------------------------------------
// Kernel 1: per (b, chunk) build W (pre-swizzled to WMMA A-operand layout)
// and P (inclusive cumprod of a). One wave per instance.
//
// A-operand layout for v_wmma_f32_16x16x4 (16x4 f32, 2 VGPRs):
//   lanes 0-15  (M=lane):    VGPR0 = K=4kk+0, VGPR1 = K=4kk+1
//   lanes 16-31 (M=lane-16): VGPR0 = K=4kk+2, VGPR1 = K=4kk+3
// Stored as Wws[inst*256 + kk*64 + lane*2 + {0,1}].
// ---------------------------------------------------------------------------
__global__ void ds_prep_kernel(const float* __restrict__ push_gate,
                               const float* __restrict__ pop_gate,
                               float* __restrict__ Wws,
                               float* __restrict__ Pws) {
  const int inst = blockIdx.x;           // b*NC + chunk
  const int b    = inst / NC;
  const int ch   = inst % NC;
  const int t0   = ch * CK;
  const int lane = threadIdx.x;

  __shared__ float a_sm[CK];
  __shared__ float c_sm[CK];

  if (lane < CK) {
    const float p  = push_gate[(size_t)b * LL + t0 + lane];
    const float o  = pop_gate [(size_t)b * LL + t0 + lane];
    const float om = 1.0f - o;
    a_sm[lane] = (1.0f - p) * om;
    c_sm[lane] = p * om;
  }
  __syncthreads();

  float* Wb = Wws + (size_t)inst * 256;

  if (lane < CK) {
    const int k = lane;                  // output row of W
    float w[CK];
    #pragma unroll
    for (int j = 0; j < CK; ++j) w[j] = 0.0f;
    w[k] = c_sm[k];
    float prod = 1.0f;
    for (int j = k - 1; j >= 0; --j) {   // suffix products of a
      prod *= a_sm[j + 1];
      w[j] = c_sm[j] * prod;
    }
    // Scatter row k into the A-operand lane layout (4 K-slices of 4).
    #pragma unroll
    for (int kk = 0; kk < 4; ++kk) {
      Wb[kk * 64 + k * 2 + 0]        = w[4 * kk + 0];
      Wb[kk * 64 + k * 2 + 1]        = w[4 * kk + 1];
      Wb[kk * 64 + (k + 16) * 2 + 0] = w[4 * kk + 2];
      Wb[kk * 64 + (k + 16) * 2 + 1] = w[4 * kk + 3];
    }
    // Inclusive cumulative product P[k] = prod_{i<=k} a[i].
    float pp = 1.0f;
    for (int i = 0; i <= k; ++i) pp *= a_sm[i];
    Pws[(size_t)inst * CK + k] = pp;
  }
}

// ---------------------------------------------------------------------------
// Kernel 2: one wave per (b, 16-wide d tile). Serial over 128 chunks:
//   acc = W @ X      (4x v_wmma_f32_16x16x4_f32)
//   acc[r] += P[row]*h ; store ; h = row 15 of acc (broadcast via shfl)
// C/D layout: lane n = column d0+(lane&15); VGPR r holds rows r (lanes 0-15)
// and r+8 (lanes 16-31). B-operand lane layout mirrors A's K striping.
// ---------------------------------------------------------------------------
__global__ void ds_scan_wmma_kernel(const float* __restrict__ x,
                                    const float* __restrict__ Wws,
                                    const float* __restrict__ Pws,
                                    float* __restrict__ out) {
  const int tile = blockIdx.x;           // 0..31  (D/16 tiles)
  const int b    = blockIdx.y;           // 0..15
  const int lane = threadIdx.x;          // 0..31
  const int half = lane >> 4;            // 0: lanes 0-15, 1: lanes 16-31
  const int n    = lane & 15;            // column within tile
  const int d0   = tile * CK;

  const float* __restrict__ xb = x   + (size_t)b * LL * DD + d0 + n;
  float*       __restrict__ ob = out + (size_t)b * LL * DD + d0 + n;
  const float* __restrict__ Wb = Wws + (size_t)b * NC * 256;
  const float* __restrict__ Pb = Pws + (size_t)b * NC * CK;

  float h = 0.0f;                        // carry h_{t0-1}[d], dup'd per half

  for (int ch = 0; ch < NC; ++ch) {
    const int t0 = ch * CK;
    const float* Wc = Wb + (size_t)ch * 256;
    const float* Pc = Pb + (size_t)ch * CK;

    v8f acc = {};
    #pragma unroll
    for (int kk = 0; kk < 4; ++kk) {
      // A slice: pre-swizzled in workspace, coalesced b64 load.
      v2f A = *(const v2f*)(Wc + kk * 64 + lane * 2);
      // B slice: X rows t0+4kk+2*half (+1), column d0+n.
      const int kb = t0 + 4 * kk + 2 * half;
      v2f Bv;
      Bv.x = xb[(size_t)kb * DD];
      Bv.y = xb[(size_t)(kb + 1) * DD];
      acc = __builtin_amdgcn_wmma_f32_16x16x4_f32(
          /*neg_a=*/false, A, /*neg_b=*/false, Bv,
          /*c_mod=*/(short)0, acc, /*reuse_a=*/false, /*reuse_b=*/false);
    }

    // Carry fixup: out[k,d] += P[k] * h_in[d].  Uniform address -> s_loads.
    float Pv[CK];
    #pragma unroll
    for (int k = 0; k < CK; ++k) Pv[k] = Pc[k];
    #pragma unroll
    for (int r = 0; r < 8; ++r) {
      const float pk = half ? Pv[r + 8] : Pv[r];   // row = r + 8*half
      acc[r] += pk * h;
    }

    // Store the 16x16 tile (row-major in out), 8 coalesced b32 stores.
    #pragma unroll
    for (int r = 0; r < 8; ++r) {
      ob[(size_t)(t0 + r + 8 * half) * DD] = acc[r];
    }

    // Next carry: row 15 lives in lanes 16-31 of acc[7]; broadcast so both
    // lane halves hold h[d0+n].
    h = __shfl(acc[7], n + 16, 32);
  }
}

// ---------------------------------------------------------------------------
// Launch. Inputs: d_in[0]=x (B,L,D) f32, d_in[1]=push_gate (B,L,1) f32,
// d_in[2]=pop_gate (B,L,1) f32. Output: (B,L,D) f32.
// Workspace layout: [ W : B*NC*256 f32 = 2 MB ][ P : B*NC*16 f32 = 128 KB ].
// ---------------------------------------------------------------------------
extern "C" void kernel_launch(void* const* d_in, const int* in_sizes, int n_in,
                              void* d_out, int out_size, void* d_ws, size_t ws_size,
                              hipStream_t stream) {
  const float* x  = (const float*)d_in[0];
  const float* pg = (const float*)d_in[1];
  const float* og = (const float*)d_in[2];
  float* out = (float*)d_out;

  float* Wws = (float*)d_ws;
  float* Pws = Wws + (size_t)BB * NC * 256;

  ds_prep_kernel<<<dim3(BB * NC), dim3(32), 0, stream>>>(pg, og, Wws, Pws);
  ds_scan_wmma_kernel<<<dim3(DD / CK, BB), dim3(32), 0, stream>>>(x, Wws, Pws, out);
}